// KGATConv_84086869721226
// MI455X (gfx1250) — compile-verified
//
#include <hip/hip_runtime.h>
#include <math.h>

#define D 64
#define NEG_SLOPE 0.01f

typedef float v2f __attribute__((ext_vector_type(2)));
typedef float v8f __attribute__((ext_vector_type(8)));

// ---- monotonic float <-> uint encoding for atomicMax on floats ----
__device__ __forceinline__ unsigned f32_to_ord(float f) {
    unsigned u = __float_as_uint(f);
    return (u & 0x80000000u) ? ~u : (u | 0x80000000u);
}
__device__ __forceinline__ float ord_to_f32(unsigned e) {
    return (e & 0x80000000u) ? __uint_as_float(e & 0x7fffffffu)
                             : __uint_as_float(~e);
}
// encoding of -inf (0xFF800000) -> ~ -> 0x007FFFFF
#define ORD_NEG_INF 0x007FFFFFu

// ---------------- Pass 0: init workspace ----------------
__global__ void kgat_init(unsigned* __restrict__ segmax,
                          float* __restrict__ denom,
                          float* __restrict__ h_nb, int N) {
    int idx = blockIdx.x * blockDim.x + threadIdx.x;
    if (idx < N) { segmax[idx] = ORD_NEG_INF; denom[idx] = 0.0f; }
    if (idx < N * D) h_nb[idx] = 0.0f;
}

// ---------------- Pass 1: fused per-edge relation matvecs + attention logits
// One wave32 per edge. Lane l owns output dims {l, l+32}. Relation weights are
// hot in L2 (512KB total); nfeat (25.6MB) is L2-resident, so src/dst gathers
// are cheap. h[src]/h[dst] broadcast via ds_bpermute (__shfl) — no LDS, no
// barriers, so per-wave early-exit guards are safe.
__global__ void kgat_edge_att(const float* __restrict__ nfeat,
                              const float* __restrict__ efeat,
                              const int* __restrict__ src,
                              const int* __restrict__ dst,
                              const int* __restrict__ etype,
                              const float* __restrict__ relW,
                              float* __restrict__ att,
                              unsigned* __restrict__ segmax, int E) {
    int lane = threadIdx.x & 31;
    int wave = threadIdx.x >> 5;
    long e = (long)blockIdx.x * 8 + wave;
    if (e >= E) return;

    int s = src[e], d = dst[e], r = etype[e];
    const float* __restrict__ hs = nfeat + (long)s * D;
    const float* __restrict__ hd = nfeat + (long)d * D;
    const float* __restrict__ W  = relW + (long)r * D * D;

    float hs_lo = hs[lane], hs_hi = hs[lane + 32];
    float hd_lo = hd[lane], hd_hi = hd[lane + 32];

    float t0 = 0.f, t1 = 0.f, r0 = 0.f, r1 = 0.f;
#pragma unroll 8
    for (int k = 0; k < 32; ++k) {
        float a = __shfl(hs_lo, k, 32);
        float b = __shfl(hd_lo, k, 32);
        float w0 = W[k * D + lane];
        float w1 = W[k * D + lane + 32];
        t0 = fmaf(a, w0, t0); t1 = fmaf(a, w1, t1);
        r0 = fmaf(b, w0, r0); r1 = fmaf(b, w1, r1);
    }
#pragma unroll 8
    for (int k = 0; k < 32; ++k) {
        float a = __shfl(hs_hi, k, 32);
        float b = __shfl(hd_hi, k, 32);
        float w0 = W[(k + 32) * D + lane];
        float w1 = W[(k + 32) * D + lane + 32];
        t0 = fmaf(a, w0, t0); t1 = fmaf(a, w1, t1);
        r0 = fmaf(b, w0, r0); r1 = fmaf(b, w1, r1);
    }

    float e0 = efeat[e * D + lane];
    float e1 = efeat[e * D + lane + 32];
    float p = t0 * tanhf(r0 + e0) + t1 * tanhf(r1 + e1);
#pragma unroll
    for (int off = 16; off > 0; off >>= 1) p += __shfl_xor(p, off, 32);

    if (lane == 0) {
        att[e] = p;
        atomicMax(&segmax[d], f32_to_ord(p));
    }
}

// ---------------- Pass 2: exp(att - max), accumulate softmax denominator
__global__ void kgat_edge_exp(const int* __restrict__ dst,
                              float* __restrict__ att,
                              const unsigned* __restrict__ segmax,
                              float* __restrict__ denom, int E) {
    long e = (long)blockIdx.x * blockDim.x + threadIdx.x;
    if (e >= E) return;
    int d = dst[e];
    float a = expf(att[e] - ord_to_f32(segmax[d]));
    att[e] = a;                       // in place: att now holds exp value
    atomicAdd(&denom[d], a);
}

// ---------------- Pass 3: weighted message scatter (u_mul_e + sum)
// 64 threads per edge, one float atomic per lane into L2-resident h_nb.
__global__ void kgat_scatter(const float* __restrict__ nfeat,
                             const int* __restrict__ src,
                             const int* __restrict__ dst,
                             const float* __restrict__ att,
                             const float* __restrict__ denom,
                             float* __restrict__ h_nb, int E) {
    long e = (long)blockIdx.x * 4 + (threadIdx.x >> 6);
    int t = threadIdx.x & 63;
    if (e >= E) return;
    int s = src[e], d = dst[e];
    float a = att[e] / denom[d];
    atomicAdd(&h_nb[(long)d * D + t], nfeat[(long)s * D + t] * a);
}

// ---------------- Pass 4: bi-interaction residual via fp32 WMMA
// out = lrelu((h + h_nb) @ W1^T) + lrelu((h * h_nb) @ W2^T)
// One wave per 16-row tile; 4 col-tiles x 2 GEMMs = 8 v8f accumulators,
// K=64 covered by 16 steps of V_WMMA_F32_16X16X4_F32 (full fp32 precision).
// A(16x4) layout: lane -> M = lane%16, K-pair = 2*(lane/16) within k-step.
// B(4x16) layout: lane -> N = lane%16, same K-pair striping.
// C/D(16x16): VGPR i -> M = i + 8*(lane/16), N = lane%16.
__global__ void kgat_residual_wmma(const float* __restrict__ nfeat,
                                   const float* __restrict__ h_nb,
                                   const float* __restrict__ w1,
                                   const float* __restrict__ w2,
                                   float* __restrict__ out, int N) {
    int lane = threadIdx.x & 31;
    int wave = threadIdx.x >> 5;
    int ntiles = (N + 15) >> 4;
    int tile = blockIdx.x * 4 + wave;      // wave-uniform: EXEC stays full
    if (tile >= ntiles) return;

    int row0 = tile << 4;
    int m = lane & 15;                     // A row / B,D column within tile
    int kb = (lane >> 4) << 1;             // K-pair select: 0 or 2
    long row = row0 + m;

    v8f acc1[4] = {};                      // (h+h_nb) @ W1^T
    v8f acc2[4] = {};                      // (h*h_nb) @ W2^T

    for (int kk = 0; kk < 16; ++kk) {
        int k0 = kk * 4 + kb;
        float hx = nfeat[row * D + k0];
        float hy = nfeat[row * D + k0 + 1];
        float nx = h_nb [row * D + k0];
        float ny = h_nb [row * D + k0 + 1];
        v2f aAdd = { hx + nx, hy + ny };
        v2f aMul = { hx * nx, hy * ny };
#pragma unroll
        for (int j = 0; j < 4; ++j) {
            int col = j * 16 + m;
            v2f b1 = { w1[col * D + k0], w1[col * D + k0 + 1] };
            v2f b2 = { w2[col * D + k0], w2[col * D + k0 + 1] };
            acc1[j] = __builtin_amdgcn_wmma_f32_16x16x4_f32(
                false, aAdd, false, b1, (short)0, acc1[j], false, false);
            acc2[j] = __builtin_amdgcn_wmma_f32_16x16x4_f32(
                false, aMul, false, b2, (short)0, acc2[j], false, false);
        }
    }

    int mtop = (lane >> 4) << 3;           // 0 or 8
#pragma unroll
    for (int j = 0; j < 4; ++j) {
#pragma unroll
        for (int i = 0; i < 8; ++i) {
            float x = acc1[j][i];
            float y = acc2[j][i];
            x = (x > 0.f) ? x : NEG_SLOPE * x;
            y = (y > 0.f) ? y : NEG_SLOPE * y;
            long rr = row0 + mtop + i;
            out[rr * D + j * 16 + m] = x + y;
        }
    }
}

extern "C" void kernel_launch(void* const* d_in, const int* in_sizes, int n_in,
                              void* d_out, int out_size, void* d_ws, size_t ws_size,
                              hipStream_t stream) {
    const float* nfeat = (const float*)d_in[0];
    const float* efeat = (const float*)d_in[1];
    const int*   src   = (const int*)d_in[2];
    const int*   dst   = (const int*)d_in[3];
    const int*   etype = (const int*)d_in[4];
    const float* relW  = (const float*)d_in[5];
    const float* w1    = (const float*)d_in[6];
    const float* w2    = (const float*)d_in[7];
    float* out = (float*)d_out;

    int N = in_sizes[0] / D;
    int E = in_sizes[2];

    // workspace layout
    unsigned* segmax = (unsigned*)d_ws;          // N u32
    float* denom = (float*)(segmax + N);         // N f32
    float* att   = denom + N;                    // E f32 (logits -> exp)
    float* h_nb  = att + E;                      // N*D f32

    // Pass 0: init
    {
        int total = N * D;
        kgat_init<<<(total + 255) / 256, 256, 0, stream>>>(segmax, denom, h_nb, N);
    }
    // Pass 1: attention logits + segment max (8 edges per 256-thread block)
    kgat_edge_att<<<(E + 7) / 8, 256, 0, stream>>>(
        nfeat, efeat, src, dst, etype, relW, att, segmax, E);
    // Pass 2: exp + denominator
    kgat_edge_exp<<<(E + 255) / 256, 256, 0, stream>>>(dst, att, segmax, denom, E);
    // Pass 3: message scatter (4 edges per 256-thread block)
    kgat_scatter<<<(E + 3) / 4, 256, 0, stream>>>(nfeat, src, dst, att, denom, h_nb, E);
    // Pass 4: fp32 WMMA residual (4 waves per block, one 16-row tile per wave)
    {
        int ntiles = (N + 15) / 16;
        kgat_residual_wmma<<<(ntiles + 3) / 4, 128, 0, stream>>>(
            nfeat, h_nb, w1, w2, out, N);
    }
}